// M50_Func_14336600834654
// MI455X (gfx1250) — compile-verified
//
#include <hip/hip_runtime.h>

// MI455X plan:
//  - I/O = 20 B/sample -> 20 MB total -> ~0.9us @ 23.3 TB/s : compute-bound.
//  - Dominant FLOPs = two 16x16 hidden layers -> V_WMMA_F32_16X16X4_F32
//    chains (full fp32 precision; K=16 done as 4 chained K=4 WMMAs).
//  - Transposed formulation h^T = W^T @ x^T keeps activations in WMMA C
//    layout; relayout C->B needs only ds_swizzle XOR-16 + cndmask (no LDS).
//  - tanh via hardware v_tanh_f32 when available (gfx1250 tanh-insts,
//    ISA lists TANH as a TRANS32 op); exp2/rcp fallback otherwise.
//  - One wave = 32 samples: lanes do per-sample interp + epilogue with full
//    EXEC (WMMA requires EXEC all-ones); only final store is predicated.

typedef float v2f __attribute__((ext_vector_type(2)));
typedef float v8f __attribute__((ext_vector_type(8)));

__device__ __forceinline__ float xor16(float x) {
  // group-of-32 swizzle, xor_mask=0x10, and_mask=0x1f: swap lane halves
  return __int_as_float(__builtin_amdgcn_ds_swizzle(__float_as_int(x), 0x401F));
}
__device__ __forceinline__ float fast_exp(float x) {
  return __builtin_amdgcn_exp2f(x * 1.44269504088896340736f);
}
__device__ __forceinline__ float fast_tanh(float x) {
#if __has_builtin(__builtin_amdgcn_tanhf)
  return __builtin_amdgcn_tanhf(x);  // v_tanh_f32: 1 TRANS op
#else
  float ax = __builtin_fabsf(x);
  float e  = __builtin_amdgcn_exp2f(ax * 2.88539008177792681472f); // 2*log2(e)
  float r  = 1.0f - 2.0f * __builtin_amdgcn_rcpf(e + 1.0f);
  return __builtin_copysignf(r, x);
#endif
}
__device__ __forceinline__ float stepf(float x) {
  return (fast_tanh(5.0f * x) + 1.0f) * 0.5f;
}
__device__ __forceinline__ v8f wmma4(v2f a, v2f b, v8f c) {
  // D = A(16x4) x B(4x16) + C(16x16), fp32
  return __builtin_amdgcn_wmma_f32_16x16x4_f32(false, a, false, b, (short)0, c,
                                               false, false);
}
__device__ __forceinline__ v8f tanh8(v8f h) {
#pragma unroll
  for (int v = 0; v < 8; ++v) h[v] = fast_tanh(h[v]);
  return h;
}

// Four 4x16 K-slices (B operands) of a 16x16 activation matrix that is held
// transposed in WMMA C layout: lane l holds rows v+8*(l>=16), col l%16.
struct BS { v2f b[4]; };
__device__ __forceinline__ BS l2b(v8f h, bool lo) {
  float s0 = xor16(h[0]), s1 = xor16(h[1]), s2 = xor16(h[2]), s3 = xor16(h[3]);
  float s4 = xor16(h[4]), s5 = xor16(h[5]), s6 = xor16(h[6]), s7 = xor16(h[7]);
  BS r;
  r.b[0].x = lo ? h[0] : s2;  r.b[0].y = lo ? h[1] : s3;   // K=0..3
  r.b[1].x = lo ? h[4] : s6;  r.b[1].y = lo ? h[5] : s7;   // K=4..7
  r.b[2].x = lo ? s0 : h[2];  r.b[2].y = lo ? s1 : h[3];   // K=8..11
  r.b[3].x = lo ? s4 : h[6];  r.b[3].y = lo ? s5 : h[7];   // K=12..15
  return r;
}

// tanh(L1) -> L2 (4 chained WMMAs) -> tanh -> L3 dot -> scalar (per sample col)
__device__ __forceinline__ float mlp23(v8f h1, const v2f a2[4], v8f c2,
                                       const float w3[8], float b3, bool lo) {
  BS bs = l2b(tanh8(h1), lo);
  v8f h2 = c2;
#pragma unroll
  for (int j = 0; j < 4; ++j) h2 = wmma4(a2[j], bs.b[j], h2);
  h2 = tanh8(h2);
  float p = 0.0f;
#pragma unroll
  for (int v = 0; v < 8; ++v) p = __builtin_fmaf(h2[v], w3[v], p);
  return b3 + p + xor16(p);  // all 32 lanes end with result for col l%16
}

__global__ __launch_bounds__(256) void hydro_wmma_kernel(
    const float* __restrict__ t, const float* __restrict__ S,
    const float* __restrict__ precp_s, const float* __restrict__ temp_s,
    const float* __restrict__ lday_s,
    const float* __restrict__ et_w1, const float* __restrict__ et_b1,
    const float* __restrict__ et_w2, const float* __restrict__ et_b2,
    const float* __restrict__ et_w3, const float* __restrict__ et_b3,
    const float* __restrict__ q_w1, const float* __restrict__ q_b1,
    const float* __restrict__ q_w2, const float* __restrict__ q_b2,
    const float* __restrict__ q_w3, const float* __restrict__ q_b3,
    const float* __restrict__ pDf, const float* __restrict__ pTmax,
    const float* __restrict__ pTmin,
    float* __restrict__ out, int B, int T) {
  const int lane = threadIdx.x & 31;
  const bool lo  = lane < 16;
  const int l16  = lane & 15;
  const int hi8  = lo ? 0 : 8;

  const float Df   = fminf(fmaxf(pDf[0], 0.01f), 5.0f);
  const float Tmax = fminf(fmaxf(pTmax[0], 0.0f), 3.0f);
  const float Tmin = fminf(fmaxf(pTmin[0], -3.0f), 0.0f);

  // ---- wave-uniform weight operands ---------------------------------------
  // Layer-1 A = W1^T as 16x4 (M=hidden, K=feature [S_snow,S_water,temp,precp])
  v2f aet1, aq1;
  aet1.x = et_w1[(lo ? 0 : 2) * 16 + l16];            // K0 / K2
  aet1.y = lo ? et_w1[1 * 16 + l16] : 0.0f;           // K1 / K3(pad)
  aq1.x  = 0.0f;                                      // K0 / K2 unused by Q
  aq1.y  = lo ? q_w1[0 * 16 + l16] : q_w1[1 * 16 + l16]; // S_water / precp
  // Layer-2 A slices = W2^T, K-slices of 4
  v2f aet2[4], aq2[4];
#pragma unroll
  for (int j = 0; j < 4; ++j) {
    int k0 = 4 * j + (lo ? 0 : 2);
    aet2[j].x = et_w2[(k0) * 16 + l16];
    aet2[j].y = et_w2[(k0 + 1) * 16 + l16];
    aq2[j].x  = q_w2[(k0) * 16 + l16];
    aq2[j].y  = q_w2[(k0 + 1) * 16 + l16];
  }
  // Biases pre-placed into the C layout (C[m][n] = b[m], m = v + hi8)
  v8f cet1, cq1, cet2, cq2;
#pragma unroll
  for (int v = 0; v < 8; ++v) {
    cet1[v] = et_b1[v + hi8];
    cq1[v]  = q_b1[v + hi8];
    cet2[v] = et_b2[v + hi8];
    cq2[v]  = q_b2[v + hi8];
  }
  float wet3[8], wq3[8];
#pragma unroll
  for (int v = 0; v < 8; ++v) { wet3[v] = et_w3[v + hi8]; wq3[v] = q_w3[v + hi8]; }
  const float bet3 = et_b3[0], bq3 = q_b3[0];

  // ---- per-lane sample: interp + feature build ----------------------------
  const int s  = blockIdx.x * blockDim.x + threadIdx.x;
  const bool valid = s < B;
  const int sc = valid ? s : (B - 1);

  float tv = t[sc];
  tv = fminf(fmaxf(tv, 0.0f), (float)(T - 1));
  int i0 = (int)tv;
  if (i0 > T - 2) i0 = T - 2;
  const float fr = tv - (float)i0;
  const float precp = precp_s[i0] + fr * (precp_s[i0 + 1] - precp_s[i0]);
  const float temp  = temp_s[i0]  + fr * (temp_s[i0 + 1]  - temp_s[i0]);
  const float lday  = lday_s[i0]  + fr * (lday_s[i0 + 1]  - lday_s[i0]);
  const float2 sv = *reinterpret_cast<const float2*>(S + 2 * sc);
  const float S_snow = sv.x, S_water = sv.y;

  // x^T (4x16) B operands: group0 = lanes 0..15's samples, group1 = 16..31's
  const float x0 = S_snow, x1 = S_water, x2 = temp, x3 = precp;
  const float sx0 = xor16(x0), sx1 = xor16(x1), sx2 = xor16(x2), sx3 = xor16(x3);
  v2f bg0, bg1;
  bg0.x = lo ? x0 : sx2;  bg0.y = lo ? x1 : sx3;
  bg1.x = lo ? sx0 : x2;  bg1.y = lo ? sx1 : x3;

  // ---- layer 1: one WMMA per (MLP, group) ---------------------------------
  v8f h_et_g0 = wmma4(aet1, bg0, cet1);
  v8f h_et_g1 = wmma4(aet1, bg1, cet1);
  v8f h_q_g0  = wmma4(aq1, bg0, cq1);
  v8f h_q_g1  = wmma4(aq1, bg1, cq1);

  // ---- layers 2+3 ----------------------------------------------------------
  const float et_g0 = mlp23(h_et_g0, aet2, cet2, wet3, bet3, lo);
  const float et_g1 = mlp23(h_et_g1, aet2, cet2, wet3, bet3, lo);
  const float q_g0  = mlp23(h_q_g0, aq2, cq2, wq3, bq3, lo);
  const float q_g1  = mlp23(h_q_g1, aq2, cq2, wq3, bq3, lo);

  const float ET = lo ? et_g0 : et_g1;  // lane l owns sample s: col l%16 of
  const float Q  = lo ? q_g0  : q_g1;   // its own group -> no shuffle needed

  // ---- bucket-model epilogue ----------------------------------------------
  const float melt = stepf(temp - Tmax) * stepf(S_snow) *
                     fminf(S_snow, Df * (temp - Tmax));
  const float dS1 = stepf(Tmin - temp) * precp - melt;
  const float sw  = stepf(S_water);
  const float dS2 = stepf(temp - Tmin) * precp + melt -
                    sw * lday * fast_exp(ET) - sw * fast_exp(Q);

  if (valid) {
    *reinterpret_cast<float2*>(out + 2 * s) = make_float2(dS1, dS2);
  }
}

extern "C" void kernel_launch(void* const* d_in, const int* in_sizes, int n_in,
                              void* d_out, int out_size, void* d_ws,
                              size_t ws_size, hipStream_t stream) {
  const float* t        = (const float*)d_in[0];
  const float* S        = (const float*)d_in[1];
  // d_in[2] = ts : uniform grid 0..T-1, folded into index math
  const float* precp_s  = (const float*)d_in[3];
  const float* temp_s   = (const float*)d_in[4];
  const float* lday_s   = (const float*)d_in[5];
  const float* et_w1 = (const float*)d_in[6];
  const float* et_b1 = (const float*)d_in[7];
  const float* et_w2 = (const float*)d_in[8];
  const float* et_b2 = (const float*)d_in[9];
  const float* et_w3 = (const float*)d_in[10];
  const float* et_b3 = (const float*)d_in[11];
  const float* q_w1  = (const float*)d_in[12];
  const float* q_b1  = (const float*)d_in[13];
  const float* q_w2  = (const float*)d_in[14];
  const float* q_b2  = (const float*)d_in[15];
  const float* q_w3  = (const float*)d_in[16];
  const float* q_b3  = (const float*)d_in[17];
  // d_in[18..20] = f, Smax, Qmax : clipped but unused by the output
  const float* pDf   = (const float*)d_in[21];
  const float* pTmax = (const float*)d_in[22];
  const float* pTmin = (const float*)d_in[23];

  const int B = in_sizes[0];
  const int T = in_sizes[2];

  const int threads = 256;
  const int blocks  = (B + threads - 1) / threads;
  hydro_wmma_kernel<<<blocks, threads, 0, stream>>>(
      t, S, precp_s, temp_s, lday_s, et_w1, et_b1, et_w2, et_b2, et_w3, et_b3,
      q_w1, q_b1, q_w2, q_b2, q_w3, q_b3, pDf, pTmax, pTmin, (float*)d_out, B,
      T);
}